// SpikingCritic_67018669686946
// MI455X (gfx1250) — compile-verified
//
#include <hip/hip_runtime.h>

typedef float v2f __attribute__((ext_vector_type(2)));
typedef float v8f __attribute__((ext_vector_type(8)));

#define Hn 128
#define Fn 16
#define Gn 4
#define Sn 20
#define WPB 4          // waves per block (one batch element per wave)
#define ROWP 20        // padded LDS row stride in floats (16 data + 4 pad, keeps 16B align)
#define LDSW (Hn*ROWP + Hn)  // per-wave LDS floats: padded W_in + I_in buffer

__device__ __forceinline__ float wsum32(float v) {
    v += __shfl_xor(v, 16, 32);
    v += __shfl_xor(v, 8, 32);
    v += __shfl_xor(v, 4, 32);
    v += __shfl_xor(v, 2, 32);
    v += __shfl_xor(v, 1, 32);
    return v;
}

__global__ __launch_bounds__(WPB * 32)
void spiking_critic_kernel(
    const float* __restrict__ energy, const float* __restrict__ threat,
    const float* __restrict__ food,   const int* __restrict__ goal,
    const int* __restrict__ prev_goal, const float* __restrict__ DA,
    const float* __restrict__ NA,     const float* __restrict__ reward,
    const float* __restrict__ cls,
    const float* __restrict__ Winw,   const float* __restrict__ Winb,
    const float* __restrict__ Woutw,  const float* __restrict__ Woutb,
    const float* __restrict__ hv0, const float* __restrict__ hu0, const float* __restrict__ hr0,
    const float* __restrict__ vv0, const float* __restrict__ vu0, const float* __restrict__ vr0,
    const float* __restrict__ elig_in, const float* __restrict__ elig_out,
    const float* __restrict__ prev_values, const float* __restrict__ prev_feat,
    const float* __restrict__ noise,
    float* __restrict__ o_values, float* __restrict__ o_td, float* __restrict__ o_act,
    float* __restrict__ o_Win, float* __restrict__ o_Wout,
    float* __restrict__ o_ein, float* __restrict__ o_eout,
    int B)
{
    __shared__ float smem[WPB * LDSW];
    const int lane = threadIdx.x & 31;
    const int wid  = threadIdx.x >> 5;
    const int b    = blockIdx.x * WPB + wid;
    if (b >= B) return;

    float* wlds = smem + wid * LDSW;      // padded W_in[b], row h at h*ROWP
    float* ilds = wlds + Hn * ROWP;       // I_in redistribution buffer (128 floats)
    const size_t bb = (size_t)b;

    // ---- Stage W_in[b] (128x16 f32) into LDS, padded rows ----
    {
        const float* gw = Winw + bb * (Hn * Fn);
        #pragma unroll
        for (int i = 0; i < 4; ++i) {
            int row = i * 32 + lane;
            const float4 x0 = *(const float4*)(gw + row * Fn + 0);
            const float4 x1 = *(const float4*)(gw + row * Fn + 4);
            const float4 x2 = *(const float4*)(gw + row * Fn + 8);
            const float4 x3 = *(const float4*)(gw + row * Fn + 12);
            float* d = wlds + row * ROWP;
            *(float4*)(d + 0)  = x0;
            *(float4*)(d + 4)  = x1;
            *(float4*)(d + 8)  = x2;
            *(float4*)(d + 12) = x3;
        }
    }

    // ---- Build feature vector (replicated in every lane) ----
    const float en = energy[b] * 0.01f;
    const int   gsel = goal[b];
    const float* cp = cls + bb * 5;
    const float da = DA[b], na = NA[b];
    const float f0 = en, f1 = threat[b], f2 = food[b], f3 = 1.0f - en;
    const float f4 = (gsel == 0) ? 1.0f : 0.0f;
    const float f5 = (gsel == 1) ? 1.0f : 0.0f;
    const float f6 = (gsel == 2) ? 1.0f : 0.0f;
    const float f7 = (gsel == 3) ? 1.0f : 0.0f;
    const float f8 = cp[0], f9 = cp[1], f10 = cp[2], f11 = cp[3], f12 = cp[4];
    const float f13 = da, f14 = na, f15 = fmaxf(0.0f, 0.5f - en);

    // ---- I_in = W_in[b] * features via V_WMMA_F32_16X16X4_F32 ----
    // A (16x4): lanes 0-15 hold M=0..15 K={0,1}; lanes 16-31 hold K={2,3}.
    // B (4x16): feature value replicated across all 16 columns -> every D column == I_in chunk.
    const int half = lane >> 4;       // 0: K lo pair, 1: K hi pair
    const int mlo  = lane & 15;       // row within 16-row chunk
    const v2f bv0 = { half ? f2  : f0,  half ? f3  : f1  };
    const v2f bv1 = { half ? f6  : f4,  half ? f7  : f5  };
    const v2f bv2 = { half ? f10 : f8,  half ? f11 : f9  };
    const v2f bv3 = { half ? f14 : f12, half ? f15 : f13 };

    #pragma unroll
    for (int r = 0; r < 8; ++r) {
        const float* arow = wlds + (r * 16 + mlo) * ROWP + 2 * half;
        const v2f a0 = *(const v2f*)(arow + 0);
        const v2f a1 = *(const v2f*)(arow + 4);
        const v2f a2 = *(const v2f*)(arow + 8);
        const v2f a3 = *(const v2f*)(arow + 12);
        v8f acc = {};
        acc = __builtin_amdgcn_wmma_f32_16x16x4_f32(false, a0, false, bv0, (short)0, acc, false, false);
        acc = __builtin_amdgcn_wmma_f32_16x16x4_f32(false, a1, false, bv1, (short)0, acc, false, false);
        acc = __builtin_amdgcn_wmma_f32_16x16x4_f32(false, a2, false, bv2, (short)0, acc, false, false);
        acc = __builtin_amdgcn_wmma_f32_16x16x4_f32(false, a3, false, bv3, (short)0, acc, false, false);
        if (mlo == 0) {   // lane 0 holds M=0..7, lane 16 holds M=8..15 (columns identical)
            float* dst = ilds + r * 16 + 8 * half;
            dst[0] = acc[0]; dst[1] = acc[1]; dst[2] = acc[2]; dst[3] = acc[3];
            dst[4] = acc[4]; dst[5] = acc[5]; dst[6] = acc[6]; dst[7] = acc[7];
        }
    }

    // ---- Redistribute I_in to ownership layout h = lane + 32j, add bias, normalize ----
    float Iin[4];
    {
        const float* bi = Winb + bb * Hn;
        float sa = 0.0f;
        #pragma unroll
        for (int j = 0; j < 4; ++j) {
            const int h = lane + 32 * j;
            Iin[j] = ilds[h] + bi[h];
            sa += fabsf(Iin[j]);
        }
        sa = wsum32(sa);
        const float sc = 5.0f / (sa * (1.0f / 128.0f) + 1e-8f);
        #pragma unroll
        for (int j = 0; j < 4; ++j) Iin[j] *= sc;
    }

    // ---- Load neuron state + W_out into registers ----
    float hv[4], hu[4], hr[4], hsp[4];
    #pragma unroll
    for (int j = 0; j < 4; ++j) {
        const int h = lane + 32 * j;
        hv[j] = hv0[bb * Hn + h];
        hu[j] = hu0[bb * Hn + h];
        hr[j] = hr0[bb * Hn + h];
        hsp[j] = 0.0f;
    }
    float wo[4][4];
    #pragma unroll
    for (int g = 0; g < 4; ++g)
        #pragma unroll
        for (int j = 0; j < 4; ++j)
            wo[g][j] = Woutw[bb * (Gn * Hn) + g * Hn + lane + 32 * j];
    const float4 obv = *(const float4*)(Woutb + bb * Gn);
    const float ob[4] = { obv.x, obv.y, obv.z, obv.w };
    const float4 vvv = *(const float4*)(vv0 + bb * Gn);
    const float4 vuv = *(const float4*)(vu0 + bb * Gn);
    const float4 vrv = *(const float4*)(vr0 + bb * Gn);
    float vv[4] = { vvv.x, vvv.y, vvv.z, vvv.w };
    float vu[4] = { vuv.x, vuv.y, vuv.z, vuv.w };
    float vr[4] = { vrv.x, vrv.y, vrv.z, vrv.w };

    // ---- 20 Izhikevich substeps ----
    const float* nzb = noise + bb * Hn;
    const size_t sstride = (size_t)B * Hn;
    for (int s = 0; s < Sn; ++s) {
        const float* nz = nzb + (size_t)s * sstride;
        if (s + 1 < Sn) __builtin_prefetch(nzb + (size_t)(s + 1) * sstride + lane, 0, 1);
        float po[4] = {0.0f, 0.0f, 0.0f, 0.0f};
        #pragma unroll
        for (int j = 0; j < 4; ++j) {
            const int h = lane + 32 * j;
            const float I = Iin[j] + 0.3f * nz[h];
            const float v = hv[j], u = hu[j];
            const float vn = v + (0.04f * v * v + 5.0f * v + 140.0f - u + I);
            const float un = u + 0.02f * (0.2f * v - u);
            const float sp = (vn >= 30.0f) ? 1.0f : 0.0f;
            hv[j] = (sp > 0.0f) ? -65.0f : vn;
            hu[j] = un + 8.0f * sp;
            hr[j] = 0.9f * hr[j] + 0.1f * sp;
            hsp[j] += sp;
            po[0] += wo[0][j] * hr[j];
            po[1] += wo[1][j] * hr[j];
            po[2] += wo[2][j] * hr[j];
            po[3] += wo[3][j] * hr[j];
        }
        float og[4];
        #pragma unroll
        for (int g = 0; g < 4; ++g) og[g] = wsum32(po[g]) + ob[g];
        const float ma = (fabsf(og[0]) + fabsf(og[1]) + fabsf(og[2]) + fabsf(og[3])) * 0.25f;
        const float sc = 4.0f / (ma + 1e-8f);
        #pragma unroll
        for (int g = 0; g < 4; ++g) {
            const float I = og[g] * sc;
            const float v = vv[g], u = vu[g];
            const float vn = v + (0.04f * v * v + 5.0f * v + 140.0f - u + I);
            const float un = u + 0.02f * (0.2f * v - u);
            const float sp = (vn >= 30.0f) ? 1.0f : 0.0f;
            vv[g] = (sp > 0.0f) ? -65.0f : vn;
            vu[g] = un + 8.0f * sp;
            vr[g] = 0.9f * vr[g] + 0.1f * sp;
        }
    }

    // ---- values, td, hidden_active ----
    float val[4];
    {
        float pv[4] = {0.0f, 0.0f, 0.0f, 0.0f};
        #pragma unroll
        for (int g = 0; g < 4; ++g) {
            #pragma unroll
            for (int j = 0; j < 4; ++j) pv[g] += wo[g][j] * hr[j];
            val[g] = wsum32(pv[g]) + ob[g];
        }
    }
    const int pg = prev_goal[b];
    const float4 pvv = *(const float4*)(prev_values + bb * Gn);
    const float Vprev = (pg == 0) ? pvv.x : (pg == 1) ? pvv.y : (pg == 2) ? pvv.z : pvv.w;
    const float Vnow  = (gsel == 0) ? val[0] : (gsel == 1) ? val[1] : (gsel == 2) ? val[2] : val[3];
    const float td = reward[b] + 0.95f * Vnow - Vprev;

    float act = (hsp[0] > 0.0f) + (hsp[1] > 0.0f) + (hsp[2] > 0.0f) + (hsp[3] > 0.0f);
    act = wsum32(act) * (1.0f / 128.0f);

    if (lane == 0) {
        *(float4*)(o_values + bb * Gn) = make_float4(val[0], val[1], val[2], val[3]);
        o_td[b]  = td;
        o_act[b] = act;
    }

    // ---- eligibility + weight updates ----
    const bool  upd = fabsf(td) > 1e-3f;
    const float sc2 = 1e-3f * td * da;

    #pragma unroll
    for (int g = 0; g < 4; ++g) {
        #pragma unroll
        for (int j = 0; j < 4; ++j) {
            const size_t idx = bb * (Gn * Hn) + g * Hn + lane + 32 * j;
            const float e = 0.9f * elig_out[idx] + vr[g] * hr[j];
            o_eout[idx] = e;
            const float w = wo[g][j];
            const float wn = fminf(3.0f, fmaxf(-3.0f, w + sc2 * e));
            o_Wout[idx] = upd ? wn : w;
        }
    }

    const float4 pf0 = *(const float4*)(prev_feat + bb * Fn + 0);
    const float4 pf1 = *(const float4*)(prev_feat + bb * Fn + 4);
    const float4 pf2 = *(const float4*)(prev_feat + bb * Fn + 8);
    const float4 pf3 = *(const float4*)(prev_feat + bb * Fn + 12);
    const float4 pf[4] = { pf0, pf1, pf2, pf3 };

    #pragma unroll
    for (int j = 0; j < 4; ++j) {
        const int h = lane + 32 * j;
        const size_t base = bb * (Hn * Fn) + (size_t)h * Fn;
        const float hrj = hr[j];
        #pragma unroll
        for (int k = 0; k < 4; ++k) {
            const float4 e4 = *(const float4*)(elig_in + base + 4 * k);
            float4 en4;
            en4.x = 0.9f * e4.x + hrj * pf[k].x;
            en4.y = 0.9f * e4.y + hrj * pf[k].y;
            en4.z = 0.9f * e4.z + hrj * pf[k].z;
            en4.w = 0.9f * e4.w + hrj * pf[k].w;
            *(float4*)(o_ein + base + 4 * k) = en4;
            const float4 w4 = *(const float4*)(wlds + h * ROWP + 4 * k);
            float4 wn4;
            wn4.x = upd ? fminf(3.0f, fmaxf(-3.0f, w4.x + sc2 * en4.x)) : w4.x;
            wn4.y = upd ? fminf(3.0f, fmaxf(-3.0f, w4.y + sc2 * en4.y)) : w4.y;
            wn4.z = upd ? fminf(3.0f, fmaxf(-3.0f, w4.z + sc2 * en4.z)) : w4.z;
            wn4.w = upd ? fminf(3.0f, fmaxf(-3.0f, w4.w + sc2 * en4.w)) : w4.w;
            *(float4*)(o_Win + base + 4 * k) = wn4;
        }
    }
}

extern "C" void kernel_launch(void* const* d_in, const int* in_sizes, int n_in,
                              void* d_out, int out_size, void* d_ws, size_t ws_size,
                              hipStream_t stream) {
    const int B = in_sizes[0];
    const float* energy      = (const float*)d_in[0];
    const float* threat      = (const float*)d_in[1];
    const float* food        = (const float*)d_in[2];
    const int*   goal        = (const int*)d_in[3];
    const int*   prev_goal   = (const int*)d_in[4];
    const float* DA          = (const float*)d_in[5];
    const float* NA          = (const float*)d_in[6];
    const float* reward      = (const float*)d_in[7];
    const float* cls         = (const float*)d_in[8];
    const float* Winw        = (const float*)d_in[9];
    const float* Winb        = (const float*)d_in[10];
    const float* Woutw       = (const float*)d_in[11];
    const float* Woutb       = (const float*)d_in[12];
    const float* hv0         = (const float*)d_in[13];
    const float* hu0         = (const float*)d_in[14];
    const float* hr0         = (const float*)d_in[15];
    const float* vv0         = (const float*)d_in[16];
    const float* vu0         = (const float*)d_in[17];
    const float* vr0         = (const float*)d_in[18];
    const float* elig_in     = (const float*)d_in[19];
    const float* elig_out    = (const float*)d_in[20];
    const float* prev_values = (const float*)d_in[21];
    const float* prev_feat   = (const float*)d_in[22];
    const float* noise       = (const float*)d_in[23];

    float* out = (float*)d_out;
    float* o_values = out;                                  // [B,G]
    float* o_td     = o_values + (size_t)B * Gn;            // [B]
    float* o_act    = o_td + B;                             // [B]
    float* o_Win    = o_act + B;                            // [B,H,F]
    float* o_Wout   = o_Win + (size_t)B * Hn * Fn;          // [B,G,H]
    float* o_ein    = o_Wout + (size_t)B * Gn * Hn;         // [B,H,F]
    float* o_eout   = o_ein + (size_t)B * Hn * Fn;          // [B,G,H]

    const int blocks = (B + WPB - 1) / WPB;
    spiking_critic_kernel<<<blocks, WPB * 32, 0, stream>>>(
        energy, threat, food, goal, prev_goal, DA, NA, reward, cls,
        Winw, Winb, Woutw, Woutb, hv0, hu0, hr0, vv0, vu0, vr0,
        elig_in, elig_out, prev_values, prev_feat, noise,
        o_values, o_td, o_act, o_Win, o_Wout, o_ein, o_eout, B);
}